// QEmbedding_33174327394652
// MI455X (gfx1250) — compile-verified
//
#include <hip/hip_runtime.h>
#include <stdint.h>

// ---------------------------------------------------------------------------
// QEmbedding int4 dequant gather for MI455X (gfx1250, wave32).
//
//   out[t, d] = (((weight[id/2, d] >> (4*(id&1))) & 0xF) - 8) * scale[id/32, d]
//
// Memory-bound: ~295 MB total traffic -> ~12.7 us floor @ 23.3 TB/s.
// One 256-thread block (8 wave32 waves) per token, 16 features per lane,
// all accesses 16B-aligned b128.  Packed bytes come in via the CDNA5 async
// global->LDS DMA (ASYNCcnt) overlapped with the scale loads; output uses
// non-temporal stores so the streaming writes don't thrash L2.
// ---------------------------------------------------------------------------

typedef uint32_t u32;
typedef u32   __attribute__((ext_vector_type(4))) u32x4;
typedef int   __attribute__((ext_vector_type(4))) i32x4;
typedef float __attribute__((ext_vector_type(4))) f32x4;

// Exact parameter types of the gfx1250 async-LDS builtin (clang prints the
// global AS as "__device__" and LDS as "__shared__" in HIP diagnostics).
typedef __attribute__((address_space(1))) i32x4 g_i32x4;   // global
typedef __attribute__((address_space(3))) i32x4 s_i32x4;   // LDS

#define QE_GROUP  32
#define QE_DIM    4096
#define QE_TPB    256          // 8 wave32 waves per block
#define QE_BPT    16           // bytes (= features) per thread; 256*16 = 4096

__global__ __launch_bounds__(QE_TPB) void qembed_dequant_kernel(
    const int*      __restrict__ ids,     // [ntok]
    const uint8_t*  __restrict__ packed,  // [NUM_EMB/2, 4096] uint8 (2 rows/byte)
    const float*    __restrict__ scales,  // [NUM_EMB/32, 4096] f32
    float*          __restrict__ out,     // [ntok, 4096] f32
    int ntok)
{
    __shared__ u32x4 lds_pk[QE_TPB];      // 4 KB staging for the packed row

    const int tok = blockIdx.x;
    if (tok >= ntok) return;

    const int tid = threadIdx.x;
    const int id  = ids[tok];                       // block-uniform -> SGPR
    const int f0  = tid * QE_BPT;                   // this lane's feature base

    const uint8_t* prow = packed + (size_t)((unsigned)id >> 1) * QE_DIM + f0;
    const f32x4*   srow = (const f32x4*)(scales +
                          (size_t)((unsigned)id / QE_GROUP) * QE_DIM + f0);
    f32x4*         orow = (f32x4*)(out + (size_t)tok * QE_DIM + f0);

    u32x4 pk;
#if defined(__gfx1250__) && __has_builtin(__builtin_amdgcn_global_load_async_to_lds_b128)
    // CDNA5 async DMA: 16B/lane global -> LDS, tracked by ASYNCcnt.
    // Issued first so it runs concurrently with the four scale loads below.
    {
        i32x4* gsrc_generic = (i32x4*)(void*)const_cast<uint8_t*>(prow);
        i32x4* ldst_generic = (i32x4*)(void*)&lds_pk[tid];
        __builtin_amdgcn_global_load_async_to_lds_b128(
            (g_i32x4*)gsrc_generic,       // global source (AS1)
            (s_i32x4*)ldst_generic,       // LDS destination (AS3)
            /*offset=*/0, /*cpol=*/0);
    }
    const bool use_async = true;
#else
    const bool use_async = false;
    pk = *(const u32x4*)prow;                       // global_load_b128 fallback
#endif

    // Scale loads overlap the async packed-row fetch.
    const f32x4 s0 = srow[0];
    const f32x4 s1 = srow[1];
    const f32x4 s2 = srow[2];
    const f32x4 s3 = srow[3];

    if (use_async) {
#if defined(__gfx1250__)
        asm volatile("s_wait_asynccnt 0" ::: "memory");  // LDS data visible
#endif
#if defined(__gfx1250__) && __has_builtin(__builtin_amdgcn_global_load_async_to_lds_b128)
        pk = lds_pk[tid];                           // ds_load_b128 (own chunk)
#endif
    }

    const int shift = (id & 1) << 2;                // 0: low nibble, 4: high

#define QE_DEQ(word, s, dst)                                                  \
    do {                                                                      \
        f32x4 o;                                                              \
        o.x = (float)((int)(((word) >> (shift +  0)) & 15u) - 8) * (s).x;     \
        o.y = (float)((int)(((word) >> (shift +  8)) & 15u) - 8) * (s).y;     \
        o.z = (float)((int)(((word) >> (shift + 16)) & 15u) - 8) * (s).z;     \
        o.w = (float)((int)(((word) >> (shift + 24)) & 15u) - 8) * (s).w;     \
        __builtin_nontemporal_store(o, (dst));  /* NT: output never re-read */\
    } while (0)

    QE_DEQ(pk.x, s0, orow + 0);
    QE_DEQ(pk.y, s1, orow + 1);
    QE_DEQ(pk.z, s2, orow + 2);
    QE_DEQ(pk.w, s3, orow + 3);
#undef QE_DEQ
}

extern "C" void kernel_launch(void* const* d_in, const int* in_sizes, int n_in,
                              void* d_out, int out_size, void* d_ws, size_t ws_size,
                              hipStream_t stream) {
    (void)n_in; (void)out_size; (void)d_ws; (void)ws_size;

    const int*     ids    = (const int*)d_in[0];      // token ids [B*S]
    const uint8_t* packed = (const uint8_t*)d_in[1];  // packed int4 table
    const float*   scales = (const float*)d_in[2];    // per-group scales
    float*         out    = (float*)d_out;

    const int ntok = in_sizes[0];                     // 4 * 2048 = 8192

    qembed_dequant_kernel<<<dim3(ntok), dim3(QE_TPB), 0, stream>>>(
        ids, packed, scales, out, ntok);
}